// SpatialEncoder_27633819582524
// MI455X (gfx1250) — compile-verified
//
#include <hip/hip_runtime.h>
#include <hip/hip_bf16.h>
#include <math.h>

typedef __attribute__((ext_vector_type(2))) float v2f;
typedef __attribute__((ext_vector_type(8))) float v8f;

#define NEG_SLOPE 0.2f
#define C_DIM 128
#define HF 128
#define NH 2

// ---------------------------------------------------------------------------
// init: out = bias broadcast, m = -inf (bits), denom = 0
// ---------------------------------------------------------------------------
__global__ void gat_init_kernel(float* __restrict__ out, unsigned* __restrict__ mbits,
                                float* __restrict__ denom, const float* __restrict__ bias,
                                long long total, long long t2) {
    long long i = (long long)blockIdx.x * blockDim.x + threadIdx.x;
    if (i < total) out[i] = bias[i & (HF - 1)];
    if (i < t2) { mbits[i] = 0xFF800000u; denom[i] = 0.0f; }
}

// ---------------------------------------------------------------------------
// Node transform GEMM via V_WMMA_F32_16X16X4_F32 (fp32-preserving).
// grid.x = T/16 row tiles, grid.y selects (Wl,bl,x_l) vs (Wr,br,x_r).
// 256 threads = 8 waves; wave w computes output columns [16w, 16w+16).
// A (16x4 f32): lanes 0-15 row M=lane K={k0,k0+1}; lanes 16-31 same rows K={k0+2,k0+3}.
// B (4x16 f32): lanes 0-15 N=lane rows K={k0,k0+1}; lanes 16-31 N=lane-16 K={k0+2,k0+3}.
// D (16x16 f32): vgpr r: lanes 0-15 -> M=r, lanes 16-31 -> M=8+r.
// ---------------------------------------------------------------------------
__global__ void gat_gemm_kernel(const float* __restrict__ x,
                                const float* __restrict__ Wl, const float* __restrict__ bl,
                                const float* __restrict__ Wr, const float* __restrict__ br,
                                float* __restrict__ xl, float* __restrict__ xr) {
    const float* W   = blockIdx.y ? Wr : Wl;
    const float* bv  = blockIdx.y ? br : bl;
    float*       out = blockIdx.y ? xr : xl;

    int wave = threadIdx.x >> 5;
    int lane = threadIdx.x & 31;
    int half = lane >> 4;     // 0: K pair {0,1}, 1: K pair {2,3}
    int l16  = lane & 15;
    long long row0 = (long long)blockIdx.x * 16;
    int col = (wave << 4) + l16;

    v8f acc = {};
    const float* ap = x + (row0 + l16) * C_DIM + 2 * half;
#pragma unroll 4
    for (int k0 = 0; k0 < C_DIM; k0 += 4) {
        v2f a = *(const v2f*)(ap + k0);          // 8B-aligned (even k)
        int ka = k0 + 2 * half;
        v2f b;
        b.x = W[(long long)ka * HF + col];
        b.y = W[(long long)(ka + 1) * HF + col];
        acc = __builtin_amdgcn_wmma_f32_16x16x4_f32(
            /*neg_a=*/false, a, /*neg_b=*/false, b,
            /*c_mod=*/(short)0, acc, /*reuse_a=*/false, /*reuse_b=*/false);
    }

    float bb = bv[col];
#pragma unroll
    for (int r = 0; r < 8; ++r) {
        long long m = row0 + r + 8 * half;
        out[m * HF + col] = acc[r] + bb;
    }
}

// ---------------------------------------------------------------------------
// Pass A: wave-per-edge score.  lane covers f4 = lane*4..lane*4+3, head = lane>>4.
// logits[e][h] = sum_f att[h][f] * leakyrelu(x_l[src]+x_r[dst]); atomic segment-max.
// ---------------------------------------------------------------------------
__device__ __forceinline__ void atomic_max_float(float* addr, float val) {
    if (val >= 0.0f) atomicMax((int*)addr, __float_as_int(val));
    else             atomicMin((unsigned int*)addr, __float_as_uint(val));
}

__global__ void gat_score_kernel(const float* __restrict__ xl, const float* __restrict__ xr,
                                 const int* __restrict__ ei, const float* __restrict__ att,
                                 float* __restrict__ logits, unsigned* __restrict__ mbits,
                                 long long E, long long Et) {
    long long wid = (long long)blockIdx.x * (blockDim.x >> 5) + (threadIdx.x >> 5);
    if (wid >= Et) return;
    int lane = threadIdx.x & 31;
    long long src, dst;
    if (wid < E) { src = ei[wid]; dst = ei[E + wid]; }
    else         { src = wid - E; dst = src; }

    float4 a = *(const float4*)(xl + src * HF + lane * 4);
    float4 b = *(const float4*)(xr + dst * HF + lane * 4);
    float4 w = *(const float4*)(att + lane * 4);
    float v, s = 0.0f;
    v = a.x + b.x; v = v > 0.0f ? v : NEG_SLOPE * v; s += v * w.x;
    v = a.y + b.y; v = v > 0.0f ? v : NEG_SLOPE * v; s += v * w.y;
    v = a.z + b.z; v = v > 0.0f ? v : NEG_SLOPE * v; s += v * w.z;
    v = a.w + b.w; v = v > 0.0f ? v : NEG_SLOPE * v; s += v * w.w;

    // reduce within each 16-lane half (one head per half)
    s += __shfl_xor(s, 1, 32);
    s += __shfl_xor(s, 2, 32);
    s += __shfl_xor(s, 4, 32);
    s += __shfl_xor(s, 8, 32);

    if ((lane & 15) == 0) {
        int h = lane >> 4;
        logits[wid * NH + h] = s;
        atomic_max_float((float*)&mbits[dst * NH + h], s);
    }
}

// ---------------------------------------------------------------------------
// Pass B: thread per (edge, head): p = exp(logit - m[dst]); atomic segment-sum.
// Overwrites logits with p.
// ---------------------------------------------------------------------------
__global__ void gat_exp_kernel(const int* __restrict__ ei, const unsigned* __restrict__ mbits,
                               float* __restrict__ logits, float* __restrict__ denom,
                               long long E, long long Et) {
    long long idx = (long long)blockIdx.x * blockDim.x + threadIdx.x;
    if (idx >= Et * NH) return;
    long long e = idx >> 1;
    int h = (int)(idx & 1);
    long long dst = (e < E) ? (long long)ei[E + e] : (e - E);
    float mm = __uint_as_float(mbits[dst * NH + h]);
    float p = expf(logits[idx] - mm);
    logits[idx] = p;
    atomicAdd(&denom[dst * NH + h], p);
}

// ---------------------------------------------------------------------------
// Pass C: wave-per-edge aggregate: out[dst] += (p/denom[dst]) * x_l[src]
// ---------------------------------------------------------------------------
__global__ void gat_agg_kernel(const float* __restrict__ xl, const int* __restrict__ ei,
                               const float* __restrict__ p, const float* __restrict__ denom,
                               float* __restrict__ out, long long E, long long Et) {
    long long wid = (long long)blockIdx.x * (blockDim.x >> 5) + (threadIdx.x >> 5);
    if (wid >= Et) return;
    int lane = threadIdx.x & 31;
    long long src, dst;
    if (wid < E) { src = ei[wid]; dst = ei[E + wid]; }
    else         { src = wid - E; dst = src; }
    int h = lane >> 4;
    float alpha = p[wid * NH + h] / denom[dst * NH + h];
    float4 xv = *(const float4*)(xl + src * HF + lane * 4);
    float* o = out + dst * HF + lane * 4;
    atomicAdd(o + 0, alpha * xv.x);
    atomicAdd(o + 1, alpha * xv.y);
    atomicAdd(o + 2, alpha * xv.z);
    atomicAdd(o + 3, alpha * xv.w);
}

// ---------------------------------------------------------------------------
extern "C" void kernel_launch(void* const* d_in, const int* in_sizes, int n_in,
                              void* d_out, int out_size, void* d_ws, size_t ws_size,
                              hipStream_t stream) {
    const float* x    = (const float*)d_in[0];
    const int*   ei   = (const int*)  d_in[1];
    const float* Wl   = (const float*)d_in[2];
    const float* bl   = (const float*)d_in[3];
    const float* Wr   = (const float*)d_in[4];
    const float* br   = (const float*)d_in[5];
    const float* att  = (const float*)d_in[6];
    const float* bias = (const float*)d_in[7];
    float* out = (float*)d_out;

    long long T  = (long long)in_sizes[0] / C_DIM;   // 80,000
    long long E  = (long long)in_sizes[1] / 2;       // 1,280,000
    long long Et = E + T;                            // + self loops

    // workspace layout
    float*    xl     = (float*)d_ws;                 // T*128
    float*    xr     = xl + T * HF;                  // T*128
    float*    logits = xr + T * HF;                  // Et*2 (becomes p)
    float*    denom  = logits + Et * NH;             // T*2
    unsigned* mbits  = (unsigned*)(denom + T * NH);  // T*2

    const int blk = 256;

    long long total = T * HF;
    long long gInit = (total + blk - 1) / blk;
    gat_init_kernel<<<dim3((unsigned)gInit), blk, 0, stream>>>(out, mbits, denom, bias, total, T * NH);

    dim3 ggrid((unsigned)(T / 16), 2);
    gat_gemm_kernel<<<ggrid, 256, 0, stream>>>(x, Wl, bl, Wr, br, xl, xr);

    long long wpb = blk / 32;
    long long gEdge = (Et + wpb - 1) / wpb;
    gat_score_kernel<<<dim3((unsigned)gEdge), blk, 0, stream>>>(xl, xr, ei, att, logits, mbits, E, Et);

    long long gExp = (Et * NH + blk - 1) / blk;
    gat_exp_kernel<<<dim3((unsigned)gExp), blk, 0, stream>>>(ei, mbits, logits, denom, E, Et);

    gat_agg_kernel<<<dim3((unsigned)gEdge), blk, 0, stream>>>(xl, ei, logits, denom, out, E, Et);
}